// M_22144851378508
// MI455X (gfx1250) — compile-verified
//
#include <hip/hip_runtime.h>
#include <math.h>

// ---------------------------------------------------------------------------
// Problem constants
// ---------------------------------------------------------------------------
#define N_ELEMS 16777216
#define BLOCK   256
#define ITEMS   16
#define CHUNK   (BLOCK * ITEMS)        // 4096 elements per block
#define NCHUNK  (N_ELEMS / CHUNK)      // 4096 chunks per stream
#define PI3     1.04719755119659774615f  // pi/3

typedef __attribute__((ext_vector_type(2))) float v2f;
typedef __attribute__((ext_vector_type(8))) float v8f;

#if __has_builtin(__builtin_amdgcn_wmma_f32_16x16x4_f32)
#define USE_WMMA_SCAN 1
#else
#define USE_WMMA_SCAN 0
#endif

// ---------------------------------------------------------------------------
// 256-element inclusive block scan (f32, full precision).
// WMMA formulation: treat the 256 thread-totals as a 16x16 tile S
// (row r = elements 16r..16r+15).
//   chain 1:  Y  = S * U        (U = upper-triangular ones, K split into 4)
//   chain 2:  D  = Lstrict * T + Y   (T[k][*] = rowTotal[k] broadcast)
// D[r][c] is then the inclusive scan of element 16r+c.
// Executed by wave 0 only (EXEC all ones, as WMMA requires).
// ---------------------------------------------------------------------------
__device__ __forceinline__ void block_scan_256(float* sm) {
    __syncthreads();   // all 256 totals visible
#if USE_WMMA_SCAN
    if (threadIdx.x < 32) {
        const int L = (int)threadIdx.x;
        const int m = L & 15;   // row (A) / column (B, C/D)
        const int h = L >> 4;   // lane-half selector

        v8f acc = {0.f, 0.f, 0.f, 0.f, 0.f, 0.f, 0.f, 0.f};

        // ---- chain 1: Y = S * U ------------------------------------------
#pragma unroll
        for (int kb = 0; kb < 4; ++kb) {
            const int k0 = 4 * kb + 2 * h;           // global K column
            v2f a = { sm[16 * m + k0], sm[16 * m + k0 + 1] };     // A = S slice
            v2f b = { (k0     <= m) ? 1.f : 0.f,                  // B = U slice
                      (k0 + 1 <= m) ? 1.f : 0.f };
            acc = __builtin_amdgcn_wmma_f32_16x16x4_f32(
                false, a, false, b, (short)0, acc, false, false);
        }

        // ---- row totals t[r] = Y[r][15] (lane 15: r=0..7, lane 31: r=8..15)
        // broadcast the 8 values each lane needs for chain 2 via shuffles.
        float tv0[4], tv1[4];
#pragma unroll
        for (int kb = 0; kb < 4; ++kb) {
            {
                const int kA = 4 * kb + 0, kB = 4 * kb + 2;
                float v0 = __shfl(acc[kA & 7], (kA < 8) ? 15 : 31, 32);
                float v1 = __shfl(acc[kB & 7], (kB < 8) ? 15 : 31, 32);
                tv0[kb] = h ? v1 : v0;
            }
            {
                const int kA = 4 * kb + 1, kB = 4 * kb + 3;
                float v0 = __shfl(acc[kA & 7], (kA < 8) ? 15 : 31, 32);
                float v1 = __shfl(acc[kB & 7], (kB < 8) ? 15 : 31, 32);
                tv1[kb] = h ? v1 : v0;
            }
        }

        // ---- chain 2: D = Lstrict * T + Y --------------------------------
#pragma unroll
        for (int kb = 0; kb < 4; ++kb) {
            const int k0 = 4 * kb + 2 * h;
            v2f a = { (k0     < m) ? 1.f : 0.f,      // A = strictly-lower ones
                      (k0 + 1 < m) ? 1.f : 0.f };
            v2f b = { tv0[kb], tv1[kb] };            // B = row totals bcast
            acc = __builtin_amdgcn_wmma_f32_16x16x4_f32(
                false, a, false, b, (short)0, acc, false, false);
        }

        // ---- write inclusive results back (C/D layout) --------------------
#pragma unroll
        for (int v = 0; v < 8; ++v)
            sm[16 * (v + 8 * h) + m] = acc[v];
    }
#else
    if (threadIdx.x == 0) {              // fallback: serial (correctness only)
        float run = 0.f;
        for (int i = 0; i < BLOCK; ++i) { run += sm[i]; sm[i] = run; }
    }
#endif
    __syncthreads();
}

// ---------------------------------------------------------------------------
// Kernel 1: per-chunk reduce.  For each chunk: local theta scan, local
// sincos, emit (Ssum = chunk theta total, Usum = sum cos(la), Vsum = sum sin(la)).
// ---------------------------------------------------------------------------
__global__ void __launch_bounds__(BLOCK)
k_reduce(const float* __restrict__ vec, const float* __restrict__ vec2,
         float* __restrict__ Ssum, float* __restrict__ Usum,
         float* __restrict__ Vsum) {
    __shared__ float sm[BLOCK];
    const int s   = blockIdx.y;
    const int c   = blockIdx.x;
    const int tid = (int)threadIdx.x;
    const float* src = (s == 0 ? vec : vec2) + (size_t)c * CHUNK + tid * ITEMS;
    const float4* src4 = (const float4*)src;

    float la[ITEMS];
    float run = 0.f;
#pragma unroll
    for (int q = 0; q < ITEMS / 4; ++q) {
        float4 xv = src4[q];
        run += PI3 * tanhf(0.5f * xv.x); la[4 * q + 0] = run;
        run += PI3 * tanhf(0.5f * xv.y); la[4 * q + 1] = run;
        run += PI3 * tanhf(0.5f * xv.z); la[4 * q + 2] = run;
        run += PI3 * tanhf(0.5f * xv.w); la[4 * q + 3] = run;
    }
    const float tot = run;

    sm[tid] = tot;
    block_scan_256(sm);
    const float excl  = sm[tid] - tot;
    const float thTot = (tid == BLOCK - 1) ? sm[tid] : 0.f;

    float u = 0.f, v = 0.f;
#pragma unroll
    for (int i = 0; i < ITEMS; ++i) {
        const float a = excl + la[i];
        u += cosf(a);
        v += sinf(a);
    }

    const int g = s * NCHUNK + c;
    sm[tid] = u;  block_scan_256(sm);
    if (tid == BLOCK - 1) Usum[g] = sm[tid];
    sm[tid] = v;  block_scan_256(sm);
    if (tid == BLOCK - 1) { Vsum[g] = sm[tid]; Ssum[g] = thTot; }
}

// ---------------------------------------------------------------------------
// Kernel 2: scan the 4096 chunk records of one stream (one block per stream).
// ang_off_c = theh + exclusive-prefix(S);  W_c = R(ang_off_c)*(U_c,V_c);
// pos_off_c = P + dl * exclusive-prefix(W).
// ---------------------------------------------------------------------------
__global__ void __launch_bounds__(BLOCK)
k_mid(const float* __restrict__ Ssum, const float* __restrict__ Usum,
      const float* __restrict__ Vsum,
      const float* __restrict__ theh, const float* __restrict__ theh2,
      const float* __restrict__ dlp,
      const float* __restrict__ P0, const float* __restrict__ P1,
      float* __restrict__ cosO, float* __restrict__ sinO,
      float* __restrict__ pxO, float* __restrict__ pyO) {
    __shared__ float sm[BLOCK];
    const int s   = blockIdx.x;
    const int tid = (int)threadIdx.x;
    const float th0 = (s == 0) ? theh[0] : theh2[0];
    const float Px  = (s == 0) ? P0[0]  : P1[0];
    const float Py  = (s == 0) ? P0[1]  : P1[1];
    const float dl  = dlp[0];
    const int base  = s * NCHUNK + tid * ITEMS;

    float laS[ITEMS];
    float run = 0.f;
#pragma unroll
    for (int i = 0; i < ITEMS; ++i) { run += Ssum[base + i]; laS[i] = run; }
    sm[tid] = run;
    block_scan_256(sm);
    const float exclS = sm[tid] - run;

    float co[ITEMS], so[ITEMS], wx[ITEMS], wy[ITEMS];
    float rx = 0.f, ry = 0.f;
#pragma unroll
    for (int i = 0; i < ITEMS; ++i) {
        const float e = th0 + exclS + (i ? laS[i - 1] : 0.f);
        co[i] = cosf(e);
        so[i] = sinf(e);
        const float U = Usum[base + i];
        const float V = Vsum[base + i];
        const float Wx = co[i] * U - so[i] * V;
        const float Wy = so[i] * U + co[i] * V;
        wx[i] = rx; rx += Wx;           // exclusive prefixes within thread
        wy[i] = ry; ry += Wy;
    }
    sm[tid] = rx;  block_scan_256(sm);  const float exX = sm[tid] - rx;
    sm[tid] = ry;  block_scan_256(sm);  const float exY = sm[tid] - ry;

#pragma unroll
    for (int i = 0; i < ITEMS; ++i) {
        const int g = base + i;
        cosO[g] = co[i];
        sinO[g] = so[i];
        pxO[g]  = Px + dl * (exX + wx[i]);
        pyO[g]  = Py + dl * (exY + wy[i]);
    }
}

// ---------------------------------------------------------------------------
// Kernel 3: downsweep.  Recompute local scans, rotate by the chunk offset,
// add chunk position offset, write interleaved [n+1, 2] rows.
// ---------------------------------------------------------------------------
__global__ void __launch_bounds__(BLOCK)
k_down(const float* __restrict__ vec, const float* __restrict__ vec2,
       const float* __restrict__ cosO, const float* __restrict__ sinO,
       const float* __restrict__ pxO, const float* __restrict__ pyO,
       const float* __restrict__ dlp,
       const float* __restrict__ P0, const float* __restrict__ P1,
       float* __restrict__ out) {
    __shared__ float sm[BLOCK];
    const int s   = blockIdx.y;
    const int c   = blockIdx.x;
    const int tid = (int)threadIdx.x;
    const float* src = (s == 0 ? vec : vec2) + (size_t)c * CHUNK + tid * ITEMS;
    const float4* src4 = (const float4*)src;

    float la[ITEMS];
    float run = 0.f;
#pragma unroll
    for (int q = 0; q < ITEMS / 4; ++q) {
        float4 xv = src4[q];
        run += PI3 * tanhf(0.5f * xv.x); la[4 * q + 0] = run;
        run += PI3 * tanhf(0.5f * xv.y); la[4 * q + 1] = run;
        run += PI3 * tanhf(0.5f * xv.z); la[4 * q + 2] = run;
        run += PI3 * tanhf(0.5f * xv.w); la[4 * q + 3] = run;
    }
    const float tot = run;
    sm[tid] = tot;
    block_scan_256(sm);
    const float excl = sm[tid] - tot;

    float cx[ITEMS], cy[ITEMS];
    float rx = 0.f, ry = 0.f;
#pragma unroll
    for (int i = 0; i < ITEMS; ++i) {
        const float a = excl + la[i];
        rx += cosf(a); cx[i] = rx;      // inclusive local step prefixes
        ry += sinf(a); cy[i] = ry;
    }
    sm[tid] = rx;  block_scan_256(sm);  const float exX = sm[tid] - rx;
    sm[tid] = ry;  block_scan_256(sm);  const float exY = sm[tid] - ry;

    const int   g  = s * NCHUNK + c;
    const float co = cosO[g], so = sinO[g];
    const float px = pxO[g],  py = pyO[g];
    const float dl = dlp[0];

    float* outS = out + (size_t)s * 2 * (N_ELEMS + 1);
    const size_t idx0 = (size_t)c * CHUNK + (size_t)tid * ITEMS;
    float2* o2 = (float2*)(outS + 2 * (idx0 + 1));
#pragma unroll
    for (int i = 0; i < ITEMS; ++i) {
        const float X = exX + cx[i];
        const float Y = exY + cy[i];
        float2 o;
        o.x = px + dl * (co * X - so * Y);
        o.y = py + dl * (so * X + co * Y);
        o2[i] = o;
    }
    if (c == 0 && tid == 0) {           // header row P[0] = P0 (resp. P1)
        outS[0] = (s == 0) ? P0[0] : P1[0];
        outS[1] = (s == 0) ? P0[1] : P1[1];
    }
}

// ---------------------------------------------------------------------------
// Launcher
// ---------------------------------------------------------------------------
extern "C" void kernel_launch(void* const* d_in, const int* in_sizes, int n_in,
                              void* d_out, int out_size, void* d_ws, size_t ws_size,
                              hipStream_t stream) {
    const float* vec   = (const float*)d_in[0];
    const float* vec2  = (const float*)d_in[1];
    const float* theh  = (const float*)d_in[2];
    const float* theh2 = (const float*)d_in[3];
    const float* dl    = (const float*)d_in[4];
    const float* P0    = (const float*)d_in[5];
    const float* P1    = (const float*)d_in[6];
    float* out = (float*)d_out;

    float* ws   = (float*)d_ws;           // 14 * NCHUNK floats = 224 KiB
    float* Ssum = ws;
    float* Usum = ws + 2 * NCHUNK;
    float* Vsum = ws + 4 * NCHUNK;
    float* cosO = ws + 6 * NCHUNK;
    float* sinO = ws + 8 * NCHUNK;
    float* pxO  = ws + 10 * NCHUNK;
    float* pyO  = ws + 12 * NCHUNK;

    dim3 grid(NCHUNK, 2);
    k_reduce<<<grid, BLOCK, 0, stream>>>(vec, vec2, Ssum, Usum, Vsum);
    k_mid<<<2, BLOCK, 0, stream>>>(Ssum, Usum, Vsum, theh, theh2, dl, P0, P1,
                                   cosO, sinO, pxO, pyO);
    k_down<<<grid, BLOCK, 0, stream>>>(vec, vec2, cosO, sinO, pxO, pyO,
                                       dl, P0, P1, out);
}